// LPMaxPool2d_30331059044681
// MI455X (gfx1250) — compile-verified
//
#include <hip/hip_runtime.h>
#include <hip/hip_bf16.h>

// Problem geometry (fixed by the reference).
constexpr int B_  = 16, C_ = 64, H_ = 224, W_ = 224;
constexpr int OH_ = 112, OW_ = 112;
constexpr int NROW         = B_ * C_ * OH_;         // 114,688 output rows
constexpr int QUADS_PER_ROW = OW_ / 4;              // 28 float4 outputs per row
constexpr int NQUADS       = NROW * QUADS_PER_ROW;  // 3,211,264 work items
constexpr int NOUT         = B_ * C_ * OH_ * OW_;   // 12,845,056 per tensor
constexpr int W4           = W_ / 4;                // 56 float4 per input row

// Launch shape: 3136 blocks x 256 threads x exactly 4 grid-stride iterations.
constexpr int BLOCK  = 256;
constexpr int NBLOCK = 3136;   // NQUADS / (NBLOCK*BLOCK) == 4 exactly

// Gaussian max of (mu_a, va) and (mu_b, vb), written into (mu_a, va).
// Matches the reference's _ab_max. Transcendentals: v_rsq_f32, v_exp_f32,
// v_rcp_f32. erf via Abramowitz-Stegun 7.1.26 (|err| <= 1.5e-7), fused with
// the pdf exponential: erf(|a|/sqrt2) = 1 - poly(t) * exp(-a^2/2).
__device__ __forceinline__ void ab_max(float& mu_a, float& va,
                                       float mu_b, float vb) {
    float s     = fmaxf(va + vb, 1e-10f);
    float rinv  = __builtin_amdgcn_rsqf(s);          // v_rsq_f32
    float vavb  = s * rinv;                          // sqrt(s)
    float d     = mu_a - mu_b;
    float alpha = d * rinv;
    float ax    = fabsf(alpha) * 0.7071067811865476f;   // |alpha|/sqrt(2)
    float E     = __expf(-ax * ax);                  // v_exp_f32: exp(-alpha^2/2)
    float pdf   = 0.3989422804014327f * E;           // E / sqrt(2*pi)
    // erf(ax) = 1 - t*(a1 + t*(a2 + t*(a3 + t*(a4 + t*a5)))) * E
    float t     = __builtin_amdgcn_rcpf(fmaf(0.3275911f, ax, 1.0f)); // v_rcp_f32
    float poly  = t * fmaf(t, fmaf(t, fmaf(t, fmaf(t, 1.061405429f,
                          -1.453152027f), 1.421413741f),
                          -0.284496736f), 0.254829592f);
    float erfv  = copysignf(fmaf(-poly, E, 1.0f), alpha);  // v_bfi_b32 sign
    float cdf   = fmaf(0.5f, erfv, 0.5f);
    float mu_c  = fmaf(vavb, pdf, fmaf(d, cdf, mu_b));
    float w     = vavb * pdf;
    float vc    = (mu_a + mu_b) * w;
    vc          = fmaf(fmaf(mu_a, mu_a, va), cdf, vc);
    vc          = fmaf(fmaf(mu_b, mu_b, vb), 1.0f - cdf, vc);
    vc          = fmaf(-mu_c, mu_c, vc);
    mu_a = mu_c;
    va   = vc;
}

// One 2x2 window reduction in reference patch order (0,0)->(0,1)->(1,0)->(1,1).
__device__ __forceinline__ void window(float m00, float v00, float m01, float v01,
                                       float m10, float v10, float m11, float v11,
                                       float& om, float& ov) {
    float ma = m00, va = v00;
    ab_max(ma, va, m01, v01);
    ab_max(ma, va, m10, v10);
    ab_max(ma, va, m11, v11);
    om = ma; ov = va;
}

__global__ __launch_bounds__(BLOCK) void lpmaxpool2d_kernel(
    const float* __restrict__ m, const float* __restrict__ v,
    float* __restrict__ out_m, float* __restrict__ out_v) {

    const float4* __restrict__ m4 = reinterpret_cast<const float4*>(m);
    const float4* __restrict__ v4 = reinterpret_cast<const float4*>(v);
    float4* __restrict__ om4 = reinterpret_cast<float4*>(out_m);
    float4* __restrict__ ov4 = reinterpret_cast<float4*>(out_v);

    const int stride = gridDim.x * blockDim.x;            // 802,816
    for (int q = blockIdx.x * blockDim.x + threadIdx.x; q < NQUADS; q += stride) {
        // Decode: output row + horizontal quad (4 output pixels) in the row.
        unsigned uq  = (unsigned)q;
        unsigned row = uq / QUADS_PER_ROW;                // (b*C + c)*OH + oh
        unsigned qw  = uq - row * QUADS_PER_ROW;
        unsigned bc  = row / OH_;
        unsigned oh  = row - bc * OH_;
        int i4 = (int)((bc * H_ + 2u * oh) * W4 + 2u * qw);  // float4 idx, row 2*oh

        // Prefetch next grid-stride iteration (global_prefetch_b8).
        int qn = q + stride;
        if (qn < NQUADS) {
            unsigned uqn  = (unsigned)qn;
            unsigned rown = uqn / QUADS_PER_ROW;
            unsigned qwn  = uqn - rown * QUADS_PER_ROW;
            unsigned bcn  = rown / OH_;
            unsigned ohn  = rown - bcn * OH_;
            int j4 = (int)((bcn * H_ + 2u * ohn) * W4 + 2u * qwn);
            __builtin_prefetch(&m4[j4],      0, 3);
            __builtin_prefetch(&m4[j4 + W4], 0, 3);
            __builtin_prefetch(&v4[j4],      0, 3);
            __builtin_prefetch(&v4[j4 + W4], 0, 3);
        }

        // 8 x global_load_b128: 8 input columns x 2 rows, means + variances.
        float4 ma0 = m4[i4],     ma1 = m4[i4 + 1];            // row 2*oh
        float4 mb0 = m4[i4 + W4], mb1 = m4[i4 + W4 + 1];      // row 2*oh+1
        float4 va0 = v4[i4],     va1 = v4[i4 + 1];
        float4 vb0 = v4[i4 + W4], vb1 = v4[i4 + W4 + 1];

        float4 om, ov;
        window(ma0.x, va0.x, ma0.y, va0.y, mb0.x, vb0.x, mb0.y, vb0.y, om.x, ov.x);
        window(ma0.z, va0.z, ma0.w, va0.w, mb0.z, vb0.z, mb0.w, vb0.w, om.y, ov.y);
        window(ma1.x, va1.x, ma1.y, va1.y, mb1.x, vb1.x, mb1.y, vb1.y, om.z, ov.z);
        window(ma1.z, va1.z, ma1.w, va1.w, mb1.z, vb1.z, mb1.w, vb1.w, om.w, ov.w);

        // Quad index q maps directly to the float4 output slot:
        // out offset = row*OW + 4*qw = 4*q. 2 x global_store_b128.
        om4[q] = om;
        ov4[q] = ov;
    }
}

extern "C" void kernel_launch(void* const* d_in, const int* in_sizes, int n_in,
                              void* d_out, int out_size, void* d_ws, size_t ws_size,
                              hipStream_t stream) {
    (void)in_sizes; (void)n_in; (void)out_size; (void)d_ws; (void)ws_size;
    const float* m = (const float*)d_in[0];   // setup_inputs order: 'm', 'v'
    const float* v = (const float*)d_in[1];
    float* out_m = (float*)d_out;             // m_a flat, then v_a flat
    float* out_v = out_m + NOUT;
    lpmaxpool2d_kernel<<<NBLOCK, BLOCK, 0, stream>>>(m, v, out_m, out_v);
}